// Generative_model_64630667870458
// MI455X (gfx1250) — compile-verified
//
#include <hip/hip_runtime.h>
#include <hip/hip_bf16.h>

typedef float v2f __attribute__((ext_vector_type(2)));
typedef float v8f __attribute__((ext_vector_type(8)));

// ---------------------------------------------------------------------------
// Zero fill (graph-capture-safe replacement for memset)
// ---------------------------------------------------------------------------
__global__ void zero_f32_kernel(float* __restrict__ p, long long n) {
    long long i = (long long)blockIdx.x * blockDim.x + threadIdx.x;
    long long stride = (long long)gridDim.x * blockDim.x;
    for (; i < n; i += stride) p[i] = 0.0f;
}

// ---------------------------------------------------------------------------
// Degree accumulation (f32 atomics, L2-resident)
// ---------------------------------------------------------------------------
__global__ void degree_kernel(const int* __restrict__ e0, const int* __restrict__ e1,
                              float* __restrict__ deg_out, float* __restrict__ deg_in, int E) {
    int i = blockIdx.x * blockDim.x + threadIdx.x;
    int stride = gridDim.x * blockDim.x;
    for (; i < E; i += stride) {
        atomicAdd(deg_out + e0[i], 1.0f);
        atomicAdd(deg_in  + e1[i], 1.0f);
    }
}

// norm = clip(deg,1)^-0.5 in place
__global__ void norm_kernel(float* __restrict__ ns, float* __restrict__ nd, int n) {
    int i = blockIdx.x * blockDim.x + threadIdx.x;
    int stride = gridDim.x * blockDim.x;
    for (; i < n; i += stride) {
        ns[i] = rsqrtf(fmaxf(ns[i], 1.0f));
        nd[i] = rsqrtf(fmaxf(nd[i], 1.0f));
    }
}

// ---------------------------------------------------------------------------
// fp32 WMMA GEMM: C[nrows x dout] = A[nrows x din] @ W[din x dout]
// One wave computes a 16 x (16*CT) strip via V_WMMA_F32_16X16X4_F32.
// Per k-step: one b64 A-fragment load reused across CT column tiles; all CT
// B fragments are loaded into distinct registers BEFORE the WMMA burst so the
// compiler can clause the loads and stagger s_wait_loadcnt (loads overlap
// matrix ops instead of serializing).
// Optional epilogue: per-row scale (norm_src pre-scaling), bias, relu.
// Requires: nrows % 16 == 0, din % 4 == 0, dout % (16*CT) == 0.
// ---------------------------------------------------------------------------
template <int CT>
__global__ void wmma_gemm_f32_kernel(const float* __restrict__ A,
                                     const float* __restrict__ W,
                                     float* __restrict__ C,
                                     int nrows, int din, int dout,
                                     const float* __restrict__ rowScale,
                                     const float* __restrict__ bias,
                                     int relu) {
    const int wavesPerBlock = blockDim.x >> 5;
    const int wave = blockIdx.x * wavesPerBlock + (threadIdx.x >> 5);
    const int lane = threadIdx.x & 31;
    const int groups   = (dout >> 4) / CT;       // column-tile groups of CT
    const int rowTiles = nrows >> 4;
    if (wave >= rowTiles * groups) return;       // wave-uniform: EXEC stays all-ones

    const int rt   = wave / groups;
    const int g    = wave - rt * groups;
    const int row0 = rt << 4;
    const int col0 = (g * CT) << 4;

    // f32 16x4 A fragment: lanes 0-15 hold K=0,1; lanes 16-31 hold K=2,3 (M = lane&15)
    const int m    = lane & 15;
    const int koff = (lane >> 4) << 1;

    const float* __restrict__ arow = A + (size_t)(row0 + m) * din;

    v8f acc[CT];
#pragma unroll
    for (int ct = 0; ct < CT; ++ct) acc[ct] = (v8f){};

    for (int k = 0; k < din; k += 4) {
        // A fragment: a.x/a.y consecutive in memory -> single b64 load
        v2f a;
        a.x = arow[k + koff];
        a.y = arow[k + koff + 1];

        // Stage ALL B fragments first (distinct regs -> one load clause,
        // staggered waits), then issue the WMMA burst.
        const float* __restrict__ w0 = W + (size_t)(k + koff) * dout + col0 + m;
        const float* __restrict__ w1 = w0 + dout;
        v2f b[CT];
#pragma unroll
        for (int ct = 0; ct < CT; ++ct) {
            b[ct].x = w0[ct * 16];
            b[ct].y = w1[ct * 16];
        }
#pragma unroll
        for (int ct = 0; ct < CT; ++ct) {
            acc[ct] = __builtin_amdgcn_wmma_f32_16x16x4_f32(
                /*neg_a=*/false, a, /*neg_b=*/false, b[ct],
                /*c_mod=*/(short)0, acc[ct], /*reuse_a=*/false, /*reuse_b=*/false);
        }
    }

    // C/D 16x16 layout: VGPR r -> M = r + 8*(lane>>4), N = lane&15
    const int rbase = (lane >> 4) ? 8 : 0;
#pragma unroll
    for (int ct = 0; ct < CT; ++ct) {
        const int col  = col0 + ct * 16 + m;
        const float bv = bias ? bias[col] : 0.0f;
#pragma unroll
        for (int r = 0; r < 8; ++r) {
            const int row = row0 + r + rbase;
            float v = acc[ct][r];
            if (rowScale) v *= rowScale[row];
            v += bv;
            if (relu) v = fmaxf(v, 0.0f);
            C[(size_t)row * dout + col] = v;
        }
    }
}

// ---------------------------------------------------------------------------
// Edge scatter-add: agg[dst[e]*d + c] += P[src[e]*d + c]
// (ns already folded into P via GEMM epilogue). VEC=4 for d=128 (float4 gather),
// VEC=1 for d=32/16. shift = log2(d/VEC).
// ---------------------------------------------------------------------------
template <int VEC>
__global__ void agg_kernel(const float* __restrict__ P, float* __restrict__ agg,
                           const int* __restrict__ src, const int* __restrict__ dst,
                           int E, int d, int shift) {
    const long long total  = (long long)E << shift;
    const int perEdgeMask  = (1 << shift) - 1;
    long long t = (long long)blockIdx.x * blockDim.x + threadIdx.x;
    const long long stride = (long long)gridDim.x * blockDim.x;
    for (; t < total; t += stride) {
        const int e = (int)(t >> shift);
        const int c = ((int)t & perEdgeMask) * VEC;
        const int s = src[e];
        const int q = dst[e];
        if (VEC == 4) {
            const float4 v = *(const float4*)(P + (size_t)s * d + c);
            float* ap = agg + (size_t)q * d + c;
            atomicAdd(ap + 0, v.x);
            atomicAdd(ap + 1, v.y);
            atomicAdd(ap + 2, v.z);
            atomicAdd(ap + 3, v.w);
        } else {
            atomicAdd(agg + (size_t)q * d + c, P[(size_t)s * d + c]);
        }
    }
}

// ---------------------------------------------------------------------------
// Finalize: out = (agg * nd[row]) + bias[col], optional relu. d = 1<<shift.
// ---------------------------------------------------------------------------
__global__ void finalize_kernel(const float* __restrict__ agg, float* __restrict__ out,
                                const float* __restrict__ nd, const float* __restrict__ bias,
                                int n, int shift, int relu) {
    const long long total = (long long)n << shift;
    const int cmask = (1 << shift) - 1;
    long long i = (long long)blockIdx.x * blockDim.x + threadIdx.x;
    const long long stride = (long long)gridDim.x * blockDim.x;
    for (; i < total; i += stride) {
        const int row = (int)(i >> shift);
        const int c   = (int)i & cmask;
        float v = agg[i] * nd[row] + bias[c];
        if (relu) v = fmaxf(v, 0.0f);
        out[i] = v;
    }
}

// ---------------------------------------------------------------------------
// Row-wise log_softmax over 32 columns: one wave32 per row, shuffle reduce.
// ---------------------------------------------------------------------------
__global__ void log_softmax32_kernel(const float* __restrict__ in, float* __restrict__ out, int n) {
    const int wave = (blockIdx.x * blockDim.x + threadIdx.x) >> 5;
    const int lane = threadIdx.x & 31;
    if (wave >= n) return;
    const float v = in[(size_t)wave * 32 + lane];
    float mx = v;
#pragma unroll
    for (int o = 16; o > 0; o >>= 1) mx = fmaxf(mx, __shfl_xor(mx, o, 32));
    float s = expf(v - mx);
#pragma unroll
    for (int o = 16; o > 0; o >>= 1) s += __shfl_xor(s, o, 32);
    out[(size_t)wave * 32 + lane] = v - mx - logf(s);
}

// ---------------------------------------------------------------------------
// Per-node score halves:
//   s0[v] = <x[v], peW[0:32]>  + <aspect[v], peW[64:96]>
//   s1[v] = <x[v], peW[32:64]> + <aspect[v], peW[96:128]>
// One wave32 per node.
// ---------------------------------------------------------------------------
__global__ void node_score_kernel(const float* __restrict__ x, const float* __restrict__ aspect,
                                  const float* __restrict__ peW,
                                  float* __restrict__ s0, float* __restrict__ s1, int n) {
    const int wave = (blockIdx.x * blockDim.x + threadIdx.x) >> 5;
    const int lane = threadIdx.x & 31;
    if (wave >= n) return;
    const float xv = x[(size_t)wave * 32 + lane];
    const float av = aspect[(size_t)wave * 32 + lane];
    float t0 = xv * peW[lane]      + av * peW[64 + lane];
    float t1 = xv * peW[32 + lane] + av * peW[96 + lane];
#pragma unroll
    for (int o = 16; o > 0; o >>= 1) {
        t0 += __shfl_xor(t0, o, 32);
        t1 += __shfl_xor(t1, o, 32);
    }
    if (lane == 0) { s0[wave] = t0; s1[wave] = t1; }
}

// score = s0[e0] + s1[e1] + pe_b for pos and neg edge lists
__global__ void edge_score_kernel(const int* __restrict__ e0, const int* __restrict__ e1,
                                  const int* __restrict__ en0, const int* __restrict__ en1,
                                  const float* __restrict__ s0, const float* __restrict__ s1,
                                  const float* __restrict__ pe_b,
                                  float* __restrict__ out_pos, float* __restrict__ out_neg, int E) {
    const float pb = pe_b[0];
    long long t = (long long)blockIdx.x * blockDim.x + threadIdx.x;
    const long long stride = (long long)gridDim.x * blockDim.x;
    const long long total = 2LL * E;
    for (; t < total; t += stride) {
        if (t < E) {
            out_pos[t] = s0[e0[t]] + s1[e1[t]] + pb;
        } else {
            const int i = (int)(t - E);
            out_neg[i] = s0[en0[i]] + s1[en1[i]] + pb;
        }
    }
}

// ---------------------------------------------------------------------------
// Host-side orchestration
// ---------------------------------------------------------------------------
static inline int nblk(long long total, int threads) {
    long long b = (total + threads - 1) / threads;
    if (b < 1) b = 1;
    return (int)b;
}

extern "C" void kernel_launch(void* const* d_in, const int* in_sizes, int n_in,
                              void* d_out, int out_size, void* d_ws, size_t ws_size,
                              hipStream_t stream) {
    const float* X      = (const float*)d_in[0];   // [N,128]
    const int*   e0     = (const int*)  d_in[1];   // [E]
    const int*   e1     = (const int*)  d_in[2];   // [E]
    const int*   eneg   = (const int*)  d_in[3];   // [2,E]
    const float* W0     = (const float*)d_in[4];
    const float* b0     = (const float*)d_in[5];
    const float* W1     = (const float*)d_in[6];
    const float* b1     = (const float*)d_in[7];
    const float* W2     = (const float*)d_in[8];
    const float* b2     = (const float*)d_in[9];
    const float* W3     = (const float*)d_in[10];
    const float* b3     = (const float*)d_in[11];
    const float* xencW  = (const float*)d_in[12];
    const float* xencB  = (const float*)d_in[13];
    const float* peW    = (const float*)d_in[14];  // [128,1]
    const float* peB    = (const float*)d_in[15];  // [1]

    const int N = in_sizes[0] / 128;
    const int E = in_sizes[1];
    const int* en0 = eneg;
    const int* en1 = eneg + E;

    float* out_pos    = (float*)d_out;
    float* out_neg    = out_pos + E;
    float* aspect_out = out_neg + E;                 // [N,32] log_softmax
    float* logits_out = aspect_out + (size_t)N * 32; // [N,16]

    // Workspace carve-up (256B aligned)
    char* ws = (char*)d_ws;
    size_t off = 0;
    auto take = [&](size_t bytes) -> float* {
        float* p = (float*)(ws + off);
        off += (bytes + 255) & ~(size_t)255;
        return p;
    };
    float* ns   = take((size_t)N * 4);        // deg_out -> norm_src
    float* nd   = take((size_t)N * 4);        // deg_in  -> norm_dst
    float* bufA = take((size_t)N * 128 * 4);  // GEMM output P
    float* bufB = take((size_t)N * 128 * 4);  // aggregation accumulator
    float* bufC = take((size_t)N * 128 * 4);  // finalized hidden state
    float* s0   = take((size_t)N * 4);
    float* s1   = take((size_t)N * 4);
    (void)ws_size; (void)n_in; (void)out_size;

    const int T = 256;

    // 1) degrees -> norms
    zero_f32_kernel<<<nblk(N, T), T, 0, stream>>>(ns, N);
    zero_f32_kernel<<<nblk(N, T), T, 0, stream>>>(nd, N);
    degree_kernel<<<nblk(E, T), T, 0, stream>>>(e0, e1, ns, nd, E);
    norm_kernel<<<nblk(N, T), T, 0, stream>>>(ns, nd, N);

    // GEMM dispatcher: CT chosen so dout % (16*CT) == 0 and A reuse maximized.
    auto gemm = [&](const float* A, const float* W, float* C, int din, int dout,
                    const float* rs, const float* bias, int relu) {
        if (dout == 128) {
            const int waves = (N >> 4) * ((dout >> 4) / 4);
            wmma_gemm_f32_kernel<4><<<nblk((long long)waves * 32, T), T, 0, stream>>>(
                A, W, C, N, din, dout, rs, bias, relu);
        } else if (dout == 32) {
            const int waves = (N >> 4) * ((dout >> 4) / 2);
            wmma_gemm_f32_kernel<2><<<nblk((long long)waves * 32, T), T, 0, stream>>>(
                A, W, C, N, din, dout, rs, bias, relu);
        } else {
            const int waves = (N >> 4) * (dout >> 4);
            wmma_gemm_f32_kernel<1><<<nblk((long long)waves * 32, T), T, 0, stream>>>(
                A, W, C, N, din, dout, rs, bias, relu);
        }
    };

    // ---- Layer 0: h1 = relu(nd * Agg(ns * (X@W0)) + b0) ----
    gemm(X, W0, bufA, 128, 128, ns, nullptr, 0);
    zero_f32_kernel<<<nblk((long long)N * 128, T), T, 0, stream>>>(bufB, (long long)N * 128);
    agg_kernel<4><<<nblk((long long)E * 32, T), T, 0, stream>>>(bufA, bufB, e0, e1, E, 128, 5);
    finalize_kernel<<<nblk((long long)N * 128, T), T, 0, stream>>>(bufB, bufC, nd, b0, N, 7, 1);

    // ---- Layer 1: h2 = relu(nd * Agg(ns * (h1@W1)) + b1) ----
    gemm(bufC, W1, bufA, 128, 128, ns, nullptr, 0);
    zero_f32_kernel<<<nblk((long long)N * 128, T), T, 0, stream>>>(bufB, (long long)N * 128);
    agg_kernel<4><<<nblk((long long)E * 32, T), T, 0, stream>>>(bufA, bufB, e0, e1, E, 128, 5);
    finalize_kernel<<<nblk((long long)N * 128, T), T, 0, stream>>>(bufB, bufC, nd, b1, N, 7, 1);

    // ---- Layer 2: aspect_embed = nd * Agg(ns * (h2@W2)) + b2  (no relu, 32-wide) ----
    gemm(bufC, W2, bufA, 128, 32, ns, nullptr, 0);
    zero_f32_kernel<<<nblk((long long)N * 32, T), T, 0, stream>>>(bufB, (long long)N * 32);
    agg_kernel<1><<<nblk((long long)E * 32, T), T, 0, stream>>>(bufA, bufB, e0, e1, E, 32, 5);
    finalize_kernel<<<nblk((long long)N * 32, T), T, 0, stream>>>(bufB, bufC, nd, b2, N, 5, 0);
    // bufC[0 : N*32] = aspect_embed

    // ---- Layer 3: logits = nd * Agg(ns * (aspect_embed@W3)) + b3  (16-wide) ----
    gemm(bufC, W3, bufA, 32, 16, ns, nullptr, 0);
    zero_f32_kernel<<<nblk((long long)N * 16, T), T, 0, stream>>>(bufB, (long long)N * 16);
    agg_kernel<1><<<nblk((long long)E * 16, T), T, 0, stream>>>(bufA, bufB, e0, e1, E, 16, 4);
    finalize_kernel<<<nblk((long long)N * 16, T), T, 0, stream>>>(bufB, logits_out, nd, b3, N, 4, 0);

    // ---- aspect = log_softmax(aspect_embed) ----
    log_softmax32_kernel<<<nblk((long long)N * 32, T), T, 0, stream>>>(bufC, aspect_out, N);

    // ---- x = relu(X @ xencW + xencB)  (bias+relu fused in WMMA epilogue) ----
    gemm(X, xencW, bufA, 128, 32, nullptr, xencB, 1);

    // ---- Factored edge scores ----
    node_score_kernel<<<nblk((long long)N * 32, T), T, 0, stream>>>(bufA, aspect_out, peW, s0, s1, N);
    edge_score_kernel<<<nblk(2LL * E, T), T, 0, stream>>>(e0, e1, en0, en1, s0, s1, peB,
                                                          out_pos, out_neg, E);
}